// MotionResidualBranch_50019189129219
// MI455X (gfx1250) — compile-verified
//
#include <hip/hip_runtime.h>

// ---------------------------------------------------------------------------
// MotionResidualBranch for MI455X (gfx1250), wave32, WMMA f32 16x16x4.
//
// Algebraic fusion: everything up to the depthwise conv is linear in a
// 15-vector of per-(b,t) statistics s = {per part p: Sv(2), Sa(2), Wsum(1)}.
//   U[b,t,:] = M16 @ [s; 1]   with M16 (256 x 16) precomputed on device.
//
// Fully fused tail: each block owns a 16-row output strip. It recomputes the
// 18-row U strip from S (16-MAC dots, registers), applies the depthwise
// conv(3) in registers, stages the conv'd A-strip in LDS, runs the 256x256
// pointwise GEMM on v_wmma_f32_16x16x4_f32, and finishes with fused
// LayerNorm.  No U / Uc intermediates ever touch global memory.
// ---------------------------------------------------------------------------

#define BD   8
#define TD   1024
#define NROW (BD * TD)        // 8192
#define KJ   54
#define DD   256
#define AS_STRIDE 258         // LDS A-strip row stride (floats), conflict-free
#define SM_STRIDE 257         // LDS output-strip row stride (floats)

typedef float v2f __attribute__((ext_vector_type(2)));
typedef float v8f __attribute__((ext_vector_type(8)));

// ---------------------------------------------------------------------------
// Kernel 0: build M16 (256 x 16).  One block of 256 threads, thread d = row.
//   cols p*5+0..1 : gamma_v * WMp[d,:] @ phi_v_w(:,c)
//   cols p*5+2..3 : gamma_a * WMp[d,:] @ phi_a_w(:,c)
//   col  p*5+4    : WMp[d,:] @ (gamma_v*phi_v_b + gamma_a*phi_a_b)
//   col  15       : WM_b[d]                    (pairs with s[15] == 1)
// ---------------------------------------------------------------------------
__global__ __launch_bounds__(256) void k_prep_M16(
    const float* __restrict__ phi_v_w, const float* __restrict__ phi_v_b,
    const float* __restrict__ phi_a_w, const float* __restrict__ phi_a_b,
    const float* __restrict__ gamma_v, const float* __restrict__ gamma_a,
    const float* __restrict__ WM_w,    const float* __restrict__ WM_b,
    float* __restrict__ M16)
{
    const int d = threadIdx.x;
    const float gv = gamma_v[0];
    const float ga = gamma_a[0];
    for (int p = 0; p < 3; ++p) {
        float a0 = 0.f, a1 = 0.f, a2 = 0.f, a3 = 0.f, a4 = 0.f;
        const float* wmp = WM_w + d * (3 * DD) + p * DD;
        for (int j = 0; j < DD; ++j) {
            const float wm = wmp[j];
            a0 += wm * phi_v_w[2 * j + 0];
            a1 += wm * phi_v_w[2 * j + 1];
            a2 += wm * phi_a_w[2 * j + 0];
            a3 += wm * phi_a_w[2 * j + 1];
            a4 += wm * (gv * phi_v_b[j] + ga * phi_a_b[j]);
        }
        M16[d * 16 + p * 5 + 0] = gv * a0;
        M16[d * 16 + p * 5 + 1] = gv * a1;
        M16[d * 16 + p * 5 + 2] = ga * a2;
        M16[d * 16 + p * 5 + 3] = ga * a3;
        M16[d * 16 + p * 5 + 4] = a4;
    }
    M16[d * 16 + 15] = WM_b[d];
}

// ---------------------------------------------------------------------------
// Kernel 1: per-(b,t) joint reduction -> S (8192 x 16), wave per row.
// V[t] = X[t]-X[t-1] (X[-1]=0); A[t] = X[t]-2X[t-1]+X[t-2] (X[-1]=X[-2]=0).
// Lane handles joints k = lane and lane+32 (k < 54). Butterfly-shuffle
// reductions across the wave32 (warpSize == 32 on gfx1250).
// ---------------------------------------------------------------------------
__global__ __launch_bounds__(128) void k_reduce_joints(
    const float* __restrict__ barX, const float* __restrict__ W,
    float* __restrict__ S)
{
    const int row  = blockIdx.x * 4 + (threadIdx.x >> 5);
    const int lane = threadIdx.x & 31;
    if (row >= NROW) return;
    const int t = row & (TD - 1);

    const float* Wr = W    + (size_t)row * KJ;
    const float* X0 = barX + (size_t)row * KJ * 2;
    const float* X1 = X0 - KJ * 2;   // valid iff t >= 1
    const float* X2 = X0 - KJ * 4;   // valid iff t >= 2

    float wk[2]  = {0.f, 0.f};
    float vx[2]  = {0.f, 0.f}, vy[2] = {0.f, 0.f};
    float ax[2]  = {0.f, 0.f}, ay[2] = {0.f, 0.f};
    int   part[2] = {0, 0};
    float wloc = 0.f;

    #pragma unroll
    for (int i = 0; i < 2; ++i) {
        const int k = lane + 32 * i;
        if (k < KJ) {
            const float w  = Wr[k];
            const float x0 = X0[2 * k], y0 = X0[2 * k + 1];
            float x1 = 0.f, y1 = 0.f, x2 = 0.f, y2 = 0.f;
            if (t >= 1) { x1 = X1[2 * k]; y1 = X1[2 * k + 1]; }
            if (t >= 2) { x2 = X2[2 * k]; y2 = X2[2 * k + 1]; }
            wk[i] = w;
            vx[i] = x0 - x1;             vy[i] = y0 - y1;
            ax[i] = x0 - 2.f * x1 + x2;  ay[i] = y0 - 2.f * y1 + y2;
            part[i] = (k < 12) ? 0 : ((k < 33) ? 1 : 2);
            wloc += w;
        }
    }

    // wave-wide sum of W
    #pragma unroll
    for (int off = 16; off > 0; off >>= 1)
        wloc += __shfl_xor(wloc, off, 32);
    const float inv = 1.f / (wloc + 1e-6f);

    float acc[15];
    #pragma unroll
    for (int i = 0; i < 15; ++i) acc[i] = 0.f;

    #pragma unroll
    for (int i = 0; i < 2; ++i) {
        const int k = lane + 32 * i;
        if (k < KJ) {
            const float wn = wk[i] * inv;
            const int   p5 = part[i] * 5;
            acc[p5 + 0] += wn * vx[i];
            acc[p5 + 1] += wn * vy[i];
            acc[p5 + 2] += wn * ax[i];
            acc[p5 + 3] += wn * ay[i];
            acc[p5 + 4] += wn;
        }
    }

    #pragma unroll
    for (int i = 0; i < 15; ++i) {
        #pragma unroll
        for (int off = 16; off > 0; off >>= 1)
            acc[i] += __shfl_xor(acc[i], off, 32);
    }

    if (lane == 0) {
        float* Sr = S + row * 16;
        #pragma unroll
        for (int i = 0; i < 15; ++i) Sr[i] = acc[i];
        Sr[15] = 1.0f;
    }
}

// ---------------------------------------------------------------------------
// Kernel 2 (fused tail): per 16-row strip
//   1) stage 18x16 S slab in LDS (rows t0-1..t0+16, zeroed outside [0,T))
//   2) thread n: u[0..17] = M16[n,:] @ S-rows (U strip column, registers),
//      depthwise conv(3) in registers -> As[m][n] in LDS
//   3) WMMA f32 16x16x4: sm = As(16x256) @ pw_w^T + pw_b
//   4) fused LayerNorm over D -> out
//
// WMMA layouts (ISA 7.12.2, wave32):
//   A 16x4:  lane lr=lane%16 -> M, VGPR v -> K = v + 2*(lane>>4)
//   B 4x16:  lane lr -> N,          VGPR v -> K = v + 2*(lane>>4)
//   C 16x16: VGPR r, element (M = r + 8*(lane>>4), N = lr)
// ---------------------------------------------------------------------------
__global__ __launch_bounds__(256) void k_fused_tail(
    const float* __restrict__ S,    const float* __restrict__ M16,
    const float* __restrict__ dw_w, const float* __restrict__ dw_b,
    const float* __restrict__ pw_w, const float* __restrict__ pw_b,
    const float* __restrict__ ln_w, const float* __restrict__ ln_b,
    float* __restrict__ out)
{
    __shared__ float Ss[18 * 16];            // S slab (rows t0-1..t0+16)
    __shared__ float As[16 * AS_STRIDE];     // conv'd activation strip
    __shared__ float sm[16 * SM_STRIDE];     // GEMM output strip for LN
    __shared__ float smu[16], srs[16];

    const int row0 = blockIdx.x * 16;        // strip never crosses a batch
    const int t0   = row0 & (TD - 1);
    const int tid  = threadIdx.x;

    // ---- stage S slab (zero rows outside the valid t range) ----
    for (int e = tid; e < 18 * 16; e += 256) {
        const int rr = e >> 4;
        const int tt = t0 - 1 + rr;
        float v = 0.f;
        if (tt >= 0 && tt < TD)
            v = S[(size_t)(row0 - 1 + rr) * 16 + (e & 15)];
        Ss[e] = v;
    }
    __syncthreads();

    // ---- per-channel U column + depthwise conv, all in registers ----
    {
        const int n = tid;                   // channel 0..255
        float mrow[16];
        #pragma unroll
        for (int k = 0; k < 16; ++k) mrow[k] = M16[n * 16 + k];

        float u[18];
        #pragma unroll
        for (int rr = 0; rr < 18; ++rr) {
            float acc = 0.f;
            #pragma unroll
            for (int k = 0; k < 16; ++k) acc += mrow[k] * Ss[rr * 16 + k];
            u[rr] = acc;                     // zero when row invalid (Ss row=0)
        }
        const float w0 = dw_w[n * 3 + 0];
        const float w1 = dw_w[n * 3 + 1];
        const float w2 = dw_w[n * 3 + 2];
        const float bb = dw_b[n];
        #pragma unroll
        for (int m = 0; m < 16; ++m)
            As[m * AS_STRIDE + n] = w0 * u[m] + w1 * u[m + 1] + w2 * u[m + 2] + bb;
    }
    __syncthreads();

    // ---- WMMA GEMM: sm = As(16x256) @ pw_w^T + pw_b ----
    const int wave = tid >> 5;
    const int lane = tid & 31;
    const int lr = lane & 15, hi = lane >> 4;

    #pragma unroll
    for (int it = 0; it < 2; ++it) {
        const int col0 = (wave * 2 + it) * 16;
        const float* brow = pw_w + (size_t)(col0 + lr) * DD;
        const float* arow = As + lr * AS_STRIDE;
        v8f c = {};
        for (int k = 0; k < DD; k += 4) {
            const int ka = k + 2 * hi;
            v2f a, b;
            a.x = arow[ka];  a.y = arow[ka + 1];
            b.x = brow[ka];  b.y = brow[ka + 1];
            c = __builtin_amdgcn_wmma_f32_16x16x4_f32(
                    false, a, false, b, (short)0, c, false, false);
        }
        #pragma unroll
        for (int r = 0; r < 8; ++r) {
            const int m = r + 8 * hi;
            const int nn = col0 + lr;
            sm[m * SM_STRIDE + nn] = c[r] + pw_b[nn];
        }
    }
    __syncthreads();

    // ---- fused LayerNorm over D ----
    if (tid < 16) {
        const int m = tid;
        float s = 0.f, s2 = 0.f;
        for (int n = 0; n < DD; ++n) {
            const float v = sm[m * SM_STRIDE + n];
            s += v; s2 += v * v;
        }
        const float mu  = s * (1.f / DD);
        const float var = s2 * (1.f / DD) - mu * mu;
        smu[m] = mu;
        srs[m] = rsqrtf(var + 1e-5f);
    }
    __syncthreads();

    const int m  = tid >> 4;
    const int nb = tid & 15;
    const float mu = smu[m], rs = srs[m];
    #pragma unroll
    for (int j = 0; j < 16; ++j) {
        const int n = j * 16 + nb;
        const float v = sm[m * SM_STRIDE + n];
        out[(size_t)(row0 + m) * DD + n] = (v - mu) * rs * ln_w[n] + ln_b[n];
    }
}

// ---------------------------------------------------------------------------
extern "C" void kernel_launch(void* const* d_in, const int* in_sizes, int n_in,
                              void* d_out, int out_size, void* d_ws, size_t ws_size,
                              hipStream_t stream)
{
    const float* barX    = (const float*)d_in[0];
    const float* W       = (const float*)d_in[1];
    const float* phi_v_w = (const float*)d_in[2];
    const float* phi_v_b = (const float*)d_in[3];
    const float* phi_a_w = (const float*)d_in[4];
    const float* phi_a_b = (const float*)d_in[5];
    const float* gamma_v = (const float*)d_in[6];
    const float* gamma_a = (const float*)d_in[7];
    const float* WM_w    = (const float*)d_in[8];
    const float* WM_b    = (const float*)d_in[9];
    const float* dw_w    = (const float*)d_in[10];
    const float* dw_b    = (const float*)d_in[11];
    const float* pw_w    = (const float*)d_in[12];
    const float* pw_b    = (const float*)d_in[13];
    const float* ln_w    = (const float*)d_in[14];
    const float* ln_b    = (const float*)d_in[15];
    float* out = (float*)d_out;

    // workspace layout (floats): M16 | S
    float* M16 = (float*)d_ws;                 //  256*16
    float* S   = M16 + DD * 16;                // 8192*16
    (void)ws_size; (void)n_in; (void)in_sizes; (void)out_size;

    k_prep_M16<<<1, 256, 0, stream>>>(phi_v_w, phi_v_b, phi_a_w, phi_a_b,
                                      gamma_v, gamma_a, WM_w, WM_b, M16);
    k_reduce_joints<<<NROW / 4, 128, 0, stream>>>(barX, W, S);
    k_fused_tail<<<NROW / 16, 256, 0, stream>>>(S, M16, dw_w, dw_b,
                                                pw_w, pw_b, ln_w, ln_b, out);
}